// OnlineDMD_52072183497421
// MI455X (gfx1250) — compile-verified
//
#include <hip/hip_runtime.h>

// ---------------------------------------------------------------------------
// OnlineDMD for MI455X (gfx1250, wave32).
//  Phase 1: sequential RLS (Sherman-Morrison) -> single persistent workgroup.
//  Phase 2: 96-step linear recurrence state <- state @ A^T, fp32 GEMM chain
//           using V_WMMA_F32_16X16X4_F32 (true fp32 matrix pipe).
// ---------------------------------------------------------------------------

typedef float v2f __attribute__((ext_vector_type(2)));
typedef float v8f __attribute__((ext_vector_type(8)));

#define NDIM  512
#define BATCH 1024
#define HOR   96
#define TLEN  96
#define LAMF  0.98f

// ---------------------------------------------------------------------------
// Phase 1: RLS update. One workgroup, 1024 threads (32 waves, one WGP).
// Threads 0..511 own P-row matvec, threads 512..1023 own A-row matvec.
// Algebra: P_new @ x == Px / denom, so A_new = A + outer(y - A x, Px/denom).
// ---------------------------------------------------------------------------
__global__ __launch_bounds__(1024) void rls_update_kernel(
    const float* __restrict__ xprev, const float* __restrict__ xnext,
    float* __restrict__ A, float* __restrict__ P) {
  __shared__ float xt[NDIM];
  __shared__ float yt[NDIM];
  __shared__ float Px[NDIM];
  __shared__ float Ax[NDIM];
  __shared__ float red[NDIM];
  __shared__ float s_invden;

  const int t = threadIdx.x;
  const float inv_lam = 1.0f / LAMF;

  for (int s = 0; s < BATCH; ++s) {
    // stage x_t / y_t in LDS
    if (t < NDIM) xt[t]        = xprev[s * NDIM + t];
    else          yt[t - NDIM] = xnext[s * NDIM + (t - NDIM)];
    __syncthreads();

    // matvecs: Px = P@x (threads 0..511), Ax = A@x (threads 512..1023)
    if (t < NDIM) {
      const float* row = P + t * NDIM;
      float acc = 0.0f;
#pragma unroll 8
      for (int j = 0; j < NDIM; ++j) acc = fmaf(row[j], xt[j], acc);
      Px[t]  = acc;
      red[t] = acc * xt[t];
    } else {
      const int i = t - NDIM;
      const float* row = A + i * NDIM;
      float acc = 0.0f;
#pragma unroll 8
      for (int j = 0; j < NDIM; ++j) acc = fmaf(row[j], xt[j], acc);
      Ax[i] = acc;
    }
    __syncthreads();

    // denom = lambda + x.Px  (tree reduction in LDS)
    for (int off = NDIM / 2; off > 0; off >>= 1) {
      if (t < off) red[t] += red[t + off];
      __syncthreads();
    }
    if (t == 0) s_invden = 1.0f / (LAMF + red[0]);
    __syncthreads();
    const float invden = s_invden;

    // rank-1 updates: 256 elements of P and A per thread
#pragma unroll 4
    for (int k = 0; k < (NDIM * NDIM) / 1024; ++k) {
      const int e = t + (k << 10);
      const int i = e >> 9;
      const int j = e & (NDIM - 1);
      P[e] = (P[e] - Px[i] * Px[j] * invden) * inv_lam;
      A[e] = fmaf(yt[i] - Ax[i], Px[j] * invden, A[e]);
    }
    __syncthreads();
  }
}

// ---------------------------------------------------------------------------
// Extract state0 = x[:, :, T-1]
// ---------------------------------------------------------------------------
__global__ __launch_bounds__(256) void init_state_kernel(
    const float* __restrict__ x, float* __restrict__ s0, int total) {
  const int i = blockIdx.x * blockDim.x + threadIdx.x;
  if (i < total) s0[i] = x[i * TLEN + (TLEN - 1)];
}

// ---------------------------------------------------------------------------
// Phase 2: one recurrence step  C[b,m] = sum_k S[b,k] * A[m,k]   (S @ A^T)
// fp32 WMMA 16x16x4. Each wave computes a 32(M) x 64(N) tile:
//   2 A-operand fragments, 4 B-operand fragments, 8 WMMAs per K-step of 4.
// A-operand (16x4 f32): lanes 0-15 hold K={k0,k0+1}, lanes 16-31 K={k0+2,k0+3}
// for M = lane%16; B-operand (4x16) mirrors it per output column -> all loads
// are contiguous float2 along K for both row-major operands.
// ---------------------------------------------------------------------------
__global__ __launch_bounds__(256) void dmd_step_kernel(
    const float* __restrict__ S, const float* __restrict__ Amat,
    float* __restrict__ Snext, float* __restrict__ Out, int h) {
  const int wave = threadIdx.x >> 5;
  const int lane = threadIdx.x & 31;
  const int half = lane >> 4;   // which K-pair this lane supplies
  const int lr   = lane & 15;   // M (for A-op) / N (for B-op) index

  const int tile = blockIdx.x * 8 + wave;   // 256 wave-tiles total
  const int m0 = (tile >> 3) * 32;          // 32 batch rows per wave
  const int n0 = (tile & 7) * 64;           // 64 output cols per wave

  v8f acc[2][4];
#pragma unroll
  for (int mt = 0; mt < 2; ++mt)
#pragma unroll
    for (int nt = 0; nt < 4; ++nt) {
      v8f z = {0.f, 0.f, 0.f, 0.f, 0.f, 0.f, 0.f, 0.f};
      acc[mt][nt] = z;
    }

  const float* srow0 = S + (m0 + lr) * NDIM + half * 2;
  const float* srow1 = S + (m0 + 16 + lr) * NDIM + half * 2;
  const float* arow0 = Amat + (n0 + 0 * 16 + lr) * NDIM + half * 2;
  const float* arow1 = Amat + (n0 + 1 * 16 + lr) * NDIM + half * 2;
  const float* arow2 = Amat + (n0 + 2 * 16 + lr) * NDIM + half * 2;
  const float* arow3 = Amat + (n0 + 3 * 16 + lr) * NDIM + half * 2;

  for (int k0 = 0; k0 < NDIM; k0 += 4) {
    v2f a0 = *(const v2f*)(srow0 + k0);
    v2f a1 = *(const v2f*)(srow1 + k0);
    v2f b0 = *(const v2f*)(arow0 + k0);
    v2f b1 = *(const v2f*)(arow1 + k0);
    v2f b2 = *(const v2f*)(arow2 + k0);
    v2f b3 = *(const v2f*)(arow3 + k0);

    acc[0][0] = __builtin_amdgcn_wmma_f32_16x16x4_f32(false, a0, false, b0,
                    (short)0, acc[0][0], false, false);
    acc[1][0] = __builtin_amdgcn_wmma_f32_16x16x4_f32(false, a1, false, b0,
                    (short)0, acc[1][0], false, false);
    acc[0][1] = __builtin_amdgcn_wmma_f32_16x16x4_f32(false, a0, false, b1,
                    (short)0, acc[0][1], false, false);
    acc[1][1] = __builtin_amdgcn_wmma_f32_16x16x4_f32(false, a1, false, b1,
                    (short)0, acc[1][1], false, false);
    acc[0][2] = __builtin_amdgcn_wmma_f32_16x16x4_f32(false, a0, false, b2,
                    (short)0, acc[0][2], false, false);
    acc[1][2] = __builtin_amdgcn_wmma_f32_16x16x4_f32(false, a1, false, b2,
                    (short)0, acc[1][2], false, false);
    acc[0][3] = __builtin_amdgcn_wmma_f32_16x16x4_f32(false, a0, false, b3,
                    (short)0, acc[0][3], false, false);
    acc[1][3] = __builtin_amdgcn_wmma_f32_16x16x4_f32(false, a1, false, b3,
                    (short)0, acc[1][3], false, false);
  }

  // C/D layout: VGPR v holds M=v (lanes 0-15) and M=v+8 (lanes 16-31), N=lane%16
#pragma unroll
  for (int mt = 0; mt < 2; ++mt)
#pragma unroll
    for (int nt = 0; nt < 4; ++nt)
#pragma unroll
      for (int v = 0; v < 8; ++v) {
        const int m = m0 + mt * 16 + v + half * 8;
        const int n = n0 + nt * 16 + lr;
        const float val = acc[mt][nt][v];
        Snext[m * NDIM + n] = val;                 // next recurrence state
        Out[(m * NDIM + n) * HOR + h] = val;       // preds in (B, N, H) layout
      }
}

// ---------------------------------------------------------------------------
// Host-side orchestration (graph-capture safe: only async ops on `stream`).
// Workspace layout (floats): P(256K) | A(256K) | Sa(512K) | Sb(512K) = 6 MB
// ---------------------------------------------------------------------------
extern "C" void kernel_launch(void* const* d_in, const int* in_sizes, int n_in,
                              void* d_out, int out_size, void* d_ws, size_t ws_size,
                              hipStream_t stream) {
  (void)in_sizes; (void)n_in; (void)out_size; (void)ws_size;

  const float* xprev = (const float*)d_in[0];  // (B, N)
  const float* xnext = (const float*)d_in[1];  // (B, N)
  const float* x     = (const float*)d_in[2];  // (B, N, T)
  const float* A0    = (const float*)d_in[3];  // (N, N)
  const float* P0    = (const float*)d_in[4];  // (N, N)
  // d_in[5] = horizon (compile-time HOR = 96)

  float* ws  = (float*)d_ws;
  float* Pws = ws;
  float* Aws = Pws + NDIM * NDIM;
  float* Sa  = Aws + NDIM * NDIM;
  float* Sb  = Sa + BATCH * NDIM;
  float* out = (float*)d_out;

  // never mutate inputs: work on workspace copies of A and P
  hipMemcpyAsync(Aws, A0, (size_t)NDIM * NDIM * sizeof(float),
                 hipMemcpyDeviceToDevice, stream);
  hipMemcpyAsync(Pws, P0, (size_t)NDIM * NDIM * sizeof(float),
                 hipMemcpyDeviceToDevice, stream);

  // Phase 1: sequential RLS, single persistent workgroup
  rls_update_kernel<<<1, 1024, 0, stream>>>(xprev, xnext, Aws, Pws);

  // Phase 2: extract last time slice then 96 GEMM steps (ping-pong state)
  init_state_kernel<<<(BATCH * NDIM + 255) / 256, 256, 0, stream>>>(
      x, Sa, BATCH * NDIM);

  float* cur = Sa;
  float* nxt = Sb;
  for (int h = 0; h < HOR; ++h) {
    dmd_step_kernel<<<32, 256, 0, stream>>>(cur, Aws, nxt, out, h);
    float* tmp = cur; cur = nxt; nxt = tmp;
  }
}